// RandLANetSeg_19593640804517
// MI455X (gfx1250) — compile-verified
//
#include <hip/hip_runtime.h>
#include <math.h>

#define B_  2
#define KNN 16

typedef float v2f __attribute__((ext_vector_type(2)));
typedef float v8f __attribute__((ext_vector_type(8)));

// ---------------------------------------------------------------------------
// Segment of the K-loop: channels [0,count) of this segment, weights at
// wbase[c], activations at src[c*NP + n]. Two n-tiles (na, nb) per wave.
// Main loop is branch-free; <4 tail uses clamped loads + value selects.
// ---------------------------------------------------------------------------
__device__ __forceinline__ void gemm_seg(const float* __restrict__ wbase,
                                         const float* __restrict__ src,
                                         int count, size_t NP, int half,
                                         int na, int nb, v8f& acc0, v8f& acc1)
{
    const float* wq = wbase + 2 * half;
    const float* qa = src + na + (size_t)(2 * half) * NP;
    const float* qb = src + nb + (size_t)(2 * half) * NP;
    int k = 0;
    for (; k + 4 <= count; k += 4) {
        v2f a;  a.x  = wq[0]; a.y  = wq[1];
        v2f v0; v0.x = qa[0]; v0.y = qa[NP];
        v2f v1; v1.x = qb[0]; v1.y = qb[NP];
        acc0 = __builtin_amdgcn_wmma_f32_16x16x4_f32(false, a, false, v0,
                                                     (short)0, acc0, false, false);
        acc1 = __builtin_amdgcn_wmma_f32_16x16x4_f32(false, a, false, v1,
                                                     (short)0, acc1, false, false);
        wq += 4; qa += 4 * NP; qb += 4 * NP;
    }
    if (k < count) {
        const int c0 = k + 2 * half, c1 = c0 + 1;
        const int g0 = (c0 < count) ? c0 : count - 1;   // clamp: loads stay valid
        const int g1 = (c1 < count) ? c1 : count - 1;
        v2f a, v0, v1;
        float wx = wbase[g0], wy = wbase[g1];
        float ax = src[(size_t)g0 * NP + na], ay = src[(size_t)g1 * NP + na];
        float bx = src[(size_t)g0 * NP + nb], by = src[(size_t)g1 * NP + nb];
        a.x  = (c0 < count) ? wx : 0.0f;  a.y  = (c1 < count) ? wy : 0.0f;
        v0.x = (c0 < count) ? ax : 0.0f;  v0.y = (c1 < count) ? ay : 0.0f;
        v1.x = (c0 < count) ? bx : 0.0f;  v1.y = (c1 < count) ? by : 0.0f;
        acc0 = __builtin_amdgcn_wmma_f32_16x16x4_f32(false, a, false, v0,
                                                     (short)0, acc0, false, false);
        acc1 = __builtin_amdgcn_wmma_f32_16x16x4_f32(false, a, false, v1,
                                                     (short)0, acc1, false, false);
    }
}

// ---------------------------------------------------------------------------
// GEMM + BN + (residual) + LeakyReLU. One wave owns a 16(m) x 32(n) tile
// (two 16x16 WMMA accumulators sharing the A/weight fetches).
// X may be split across two buffers at channel `csplit` (concat elimination).
// Requires Npos % 32 == 0 (true at every call site; min Npos = 128).
// ---------------------------------------------------------------------------
__global__ __launch_bounds__(256) void wmma_conv_kernel(
    const float* __restrict__ W,
    const float* __restrict__ gamma,
    const float* __restrict__ beta,
    const float* __restrict__ in1, int i1ct, int i1co,
    const float* __restrict__ in2, int i2ct, int i2co, int csplit,
    const float* __restrict__ res, int rct, int rco,
    float* __restrict__ out, int oct, int oco,
    int Cout, int Cin, int Npos, int act)
{
    const int wave = threadIdx.x >> 5;
    const int lane = threadIdx.x & 31;
    const int n0   = (blockIdx.y * 8 + wave) * 32;
    if (n0 >= Npos) return;                 // wave-uniform exit (EXEC all-1 for WMMA)
    const int m0   = blockIdx.x * 16;
    const int b    = blockIdx.z;
    const int half = lane >> 4;
    const int l    = lane & 15;
    const int na   = n0 + l;
    const int nb   = na + 16;
    const size_t NP = (size_t)Npos;

    const int wrow = (m0 + l < Cout) ? (m0 + l) : (Cout - 1);   // clamp, no guard
    const float* __restrict__ wp = W + (size_t)wrow * (size_t)Cin;

    v8f acc0 = {}, acc1 = {};
    const int s0 = (csplit < Cin) ? csplit : Cin;

    if (s0 > 0) {
        const float* p1 = in1 + ((size_t)b * i1ct + i1co) * NP;
        gemm_seg(wp, p1, s0, NP, half, na, nb, acc0, acc1);
    }
    if (Cin > s0) {
        const float* p2 = in2 + ((size_t)b * i2ct + i2co) * NP;
        gemm_seg(wp + s0, p2, Cin - s0, NP, half, na, nb, acc0, acc1);
    }

    const float bnscale = 0.9999995000003750f;   // 1/sqrt(1 + 1e-6)
    float* __restrict__ ob = out + ((size_t)b * oct + oco) * NP;
    const float* __restrict__ rb = res ? res + ((size_t)b * rct + rco) * NP
                                       : (const float*)0;
    #pragma unroll
    for (int v = 0; v < 8; ++v) {
        const int m = m0 + v + 8 * half;
        if (m < Cout) {
            float y0 = acc0[v], y1 = acc1[v];
            if (gamma) { const float g = gamma[m] * bnscale; y0 *= g; y1 *= g; }
            if (beta)  { const float bb = beta[m]; y0 += bb; y1 += bb; }
            if (res)   { y0 += rb[(size_t)m * NP + na]; y1 += rb[(size_t)m * NP + nb]; }
            if (act)   { y0 = (y0 > 0.0f) ? y0 : 0.2f * y0;
                         y1 = (y1 > 0.0f) ? y1 : 0.2f * y1; }
            ob[(size_t)m * NP + na] = y0;
            ob[(size_t)m * NP + nb] = y1;
        }
    }
}

// relative_pos_encoding: out (B,10,N*K) = [dist, rel(3), center(3), neigh(3)]
__global__ void relpos_kernel(const float* __restrict__ xyz,
                              const int* __restrict__ nidx,
                              float* __restrict__ out, int N)
{
    const int NK = N * KNN;
    int i = blockIdx.x * 256 + threadIdx.x;
    if (i >= NK) return;
    const int b  = blockIdx.z;
    const int nn = i >> 4;
    const int id = nidx[((size_t)b * N + nn) * KNN + (i & 15)];
    const float* xb = xyz + (size_t)b * N * 3;
    const float cx = xb[3 * nn], cy = xb[3 * nn + 1], cz = xb[3 * nn + 2];
    const float px = xb[3 * id], py = xb[3 * id + 1], pz = xb[3 * id + 2];
    const float rx = cx - px, ry = cy - py, rz = cz - pz;
    const float dist = sqrtf(fmaxf(rx * rx + ry * ry + rz * rz, 1e-20f));
    const size_t s = (size_t)NK;
    float* o = out + (size_t)b * 10 * s + (size_t)i;
    o[0]     = dist;
    o[s]     = rx;  o[2 * s] = ry;  o[3 * s] = rz;
    o[4 * s] = cx;  o[5 * s] = cy;  o[6 * s] = cz;
    o[7 * s] = px;  o[8 * s] = py;  o[9 * s] = pz;
}

// gather neighbours: out (B,C,N*K); out[b,c,n*K+k] = feat[b,c,idx[b,n,k]]
__global__ void gather_nk_kernel(const float* __restrict__ feat, int fct, int fco,
                                 const int* __restrict__ nidx,
                                 float* __restrict__ out, int oct, int oco, int N)
{
    const int NK = N * KNN;
    int i = blockIdx.x * 256 + threadIdx.x;
    if (i >= NK) return;
    const int c = blockIdx.y, b = blockIdx.z;
    const int id = nidx[((size_t)b * N + (i >> 4)) * KNN + (i & 15)];
    out[((size_t)b * oct + oco + c) * (size_t)NK + i] =
        feat[((size_t)b * fct + fco + c) * (size_t)N + id];
}

// attention pooling core: per (b,c,n) softmax over K of att, weighted sum of fs
__global__ void softmax_agg_kernel(const float* __restrict__ att,
                                   const float* __restrict__ fs1, int csplit,
                                   const float* __restrict__ fs2,
                                   float* __restrict__ out, int C, int N)
{
    int n = blockIdx.x * 256 + threadIdx.x;
    if (n >= N) return;
    const int c = blockIdx.y, b = blockIdx.z;
    const size_t NK = (size_t)N * KNN;
    const float* ap = att + ((size_t)b * C + c) * NK + (size_t)n * KNN;
    const float* fp = (c < csplit)
        ? fs1 + ((size_t)b * csplit + c) * NK + (size_t)n * KNN
        : fs2 + ((size_t)b * (C - csplit) + (c - csplit)) * NK + (size_t)n * KNN;
    float mx = ap[0];
    #pragma unroll
    for (int k = 1; k < KNN; ++k) mx = fmaxf(mx, ap[k]);
    float s = 0.0f, acc = 0.0f;
    #pragma unroll
    for (int k = 0; k < KNN; ++k) {
        const float e = expf(ap[k] - mx);
        s   += e;
        acc += fp[k] * e;
    }
    out[((size_t)b * C + c) * (size_t)N + n] = acc / s;
}

// random_sample: out (B,C,M) = max_k feat[b,c,sidx[b,m,k]]
__global__ void maxpool_kernel(const float* __restrict__ feat,
                               const int* __restrict__ sidx,
                               float* __restrict__ out, int C, int N, int M)
{
    int m = blockIdx.x * 256 + threadIdx.x;
    if (m >= M) return;
    const int c = blockIdx.y, b = blockIdx.z;
    const int* ip = sidx + ((size_t)b * M + m) * KNN;
    const float* fp = feat + ((size_t)b * C + c) * (size_t)N;
    float v = fp[ip[0]];
    #pragma unroll
    for (int k = 1; k < KNN; ++k) v = fmaxf(v, fp[ip[k]]);
    out[((size_t)b * C + c) * (size_t)M + m] = v;
}

// nearest_interp: out (B,C,N) = feat[b,c,iidx[b,n]]
__global__ void interp_kernel(const float* __restrict__ feat,
                              const int* __restrict__ iidx,
                              float* __restrict__ out, int C, int M, int N)
{
    int n = blockIdx.x * 256 + threadIdx.x;
    if (n >= N) return;
    const int c = blockIdx.y, b = blockIdx.z;
    out[((size_t)b * C + c) * (size_t)N + n] =
        feat[((size_t)b * C + c) * (size_t)M + iidx[(size_t)b * N + n]];
}

// ---------------------------------------------------------------------------
extern "C" void kernel_launch(void* const* d_in, const int* in_sizes, int n_in,
                              void* d_out, int out_size, void* d_ws, size_t ws_size,
                              hipStream_t stream)
{
    (void)in_sizes; (void)n_in; (void)out_size; (void)ws_size;
    static const int NS[5] = {65536, 16384, 4096, 1024, 256};
    static const int MS[5] = {16384, 4096, 1024, 256, 128};
    static const int OC[5] = {16, 64, 128, 256, 512};

    // --- inputs (setup_inputs insertion order, interleaved per layer) ---
    const float* xyz[5]; const int* nidx[5]; const int* sidx[5]; const int* iidx[5];
    for (int i = 0; i < 5; ++i) {
        xyz[i]  = (const float*)d_in[4 * i + 0];
        nidx[i] = (const int*)  d_in[4 * i + 1];
        sidx[i] = (const int*)  d_in[4 * i + 2];
        iidx[i] = (const int*)  d_in[4 * i + 3];
    }
    const float* features = (const float*)d_in[20];

    // --- params: jax pytree flatten (dict keys sorted, lists in order) ---
    int t = 21;
    auto nxt = [&]() { return (const float*)d_in[t++]; };
    struct ConvW { const float *bta, *g, *w; };
    auto getConv = [&]() { ConvW c; c.bta = nxt(); c.g = nxt(); c.w = nxt(); return c; };

    ConvW decW[5];                                   // 'dec'
    for (int j = 0; j < 5; ++j) decW[j] = getConv();
    ConvW dec0W = getConv();                         // 'dec0'
    struct EncW { const float* a1fc; ConvW a1m; const float* a2fc; ConvW a2m;
                  ConvW lm1, lm2, m1, m2, sc; };
    EncW encW[5];                                    // 'enc'
    for (int i = 0; i < 5; ++i) {
        encW[i].a1fc = nxt(); encW[i].a1m = getConv();   // lfa.att1
        encW[i].a2fc = nxt(); encW[i].a2m = getConv();   // lfa.att2
        encW[i].lm1  = getConv();                        // lfa.mlp1
        encW[i].lm2  = getConv();                        // lfa.mlp2
        encW[i].m1   = getConv();                        // mlp1
        encW[i].m2   = getConv();                        // mlp2
        encW[i].sc   = getConv();                        // shortcut
    }
    ConvW fc0W = getConv(), fc1W = getConv(), fc2W = getConv();  // fc0,fc1,fc2
    const float* fc3b = nxt(); const float* fc3w = nxt();        // fc3: b, w

    // --- GEMM launcher ---
    auto conv = [&](const float* w, const float* g, const float* bta,
                    int Cout, int Cin,
                    const float* in1, int i1ct, int i1co, int csplit,
                    const float* in2, int i2ct, int i2co,
                    const float* res, int rct, int rco,
                    float* out, int oct, int oco, int Npos, int act) {
        dim3 grid((Cout + 15) / 16, (Npos + 255) / 256, B_);
        hipLaunchKernelGGL(wmma_conv_kernel, grid, dim3(256), 0, stream,
                           w, g, bta, in1, i1ct, i1co, in2, i2ct, i2co, csplit,
                           res, rct, rco, out, oct, oco, Cout, Cin, Npos, act);
    };

    // --- workspace arena (floats) ---
    float* ws = (float*)d_ws;
    size_t off = 0;
    auto alloc = [&](size_t nf) { float* p = ws + off; off += nf; return p; };
    float* skipFE0 = alloc((size_t)B_ * 32 * 65536);             // enc[0] = fe0
    float* skipP[5];
    for (int i = 0; i < 5; ++i) skipP[i] = alloc((size_t)B_ * 2 * OC[i] * MS[i]);
    float* f0buf = alloc((size_t)B_ * 8 * 65536);                // fc0 output
    const size_t sbase = off;                                    // reusable scratch

    // fc0: 6 -> 8 (+BN+leaky)
    conv(fc0W.w, fc0W.g, fc0W.bta, 8, 6,
         features, 6, 0, 6, nullptr, 0, 0, nullptr, 0, 0,
         f0buf, 8, 0, NS[0], 1);

    // ----------------- encoder -----------------
    for (int i = 0; i < 5; ++i) {
        const int N = NS[i], NK = N * KNN, d = OC[i], dh = d / 2;
        const int din = (i == 0) ? 8 : 2 * OC[i - 1];
        const float* fin = (i == 0) ? f0buf : skipP[i - 1];
        size_t so = sbase;
        auto sal = [&](size_t nf) { float* p = ws + so; so += nf; return p; };
        float* f1   = sal((size_t)B_ * dh * N);
        float* fn   = sal((size_t)B_ * dh * NK);
        float* fxyz = sal((size_t)B_ * dh * NK);
        float* ovl  = sal((size_t)B_ * ((10 > dh) ? 10 : dh) * NK);  // pos10 then fxyz2
        float* attb = sal((size_t)B_ * d * NK);
        float* aggb = sal((size_t)B_ * d * N);
        float* fagg = sal((size_t)B_ * dh * N);
        float* f2b  = sal((size_t)B_ * d * N);
        float* scb  = sal((size_t)B_ * 2 * d * N);
        float* feb  = (i == 0) ? skipFE0 : sal((size_t)B_ * 2 * d * N);
        const EncW& E = encW[i];

        // mlp1: din -> dh
        conv(E.m1.w, E.m1.g, E.m1.bta, dh, din, fin, din, 0, din,
             nullptr, 0, 0, nullptr, 0, 0, f1, dh, 0, N, 1);
        // relative position encoding -> ovl (10 ch)
        { dim3 g((NK + 255) / 256, 1, B_);
          hipLaunchKernelGGL(relpos_kernel, g, dim3(256), 0, stream, xyz[i], nidx[i], ovl, N); }
        // lfa.mlp1: 10 -> dh over N*K
        conv(E.lm1.w, E.lm1.g, E.lm1.bta, dh, 10, ovl, 10, 0, 10,
             nullptr, 0, 0, nullptr, 0, 0, fxyz, dh, 0, NK, 1);
        // gather f1 -> fn
        { dim3 g((NK + 255) / 256, dh, B_);
          hipLaunchKernelGGL(gather_nk_kernel, g, dim3(256), 0, stream, f1, dh, 0, nidx[i], fn, dh, 0, N); }
        // att1 fc (d x d), split input [fn | fxyz]
        conv(E.a1fc, nullptr, nullptr, d, d, fn, dh, 0, dh, fxyz, dh, 0,
             nullptr, 0, 0, attb, d, 0, NK, 0);
        { dim3 g((N + 255) / 256, d, B_);
          hipLaunchKernelGGL(softmax_agg_kernel, g, dim3(256), 0, stream, attb, fn, dh, fxyz, aggb, d, N); }
        // att1 mlp: d -> dh
        conv(E.a1m.w, E.a1m.g, E.a1m.bta, dh, d, aggb, d, 0, d,
             nullptr, 0, 0, nullptr, 0, 0, fagg, dh, 0, N, 1);
        // lfa.mlp2: dh -> dh on fxyz, output into overlay slot
        conv(E.lm2.w, E.lm2.g, E.lm2.bta, dh, dh, fxyz, dh, 0, dh,
             nullptr, 0, 0, nullptr, 0, 0, ovl, dh, 0, NK, 1);
        // gather fagg -> fn (reuse)
        { dim3 g((NK + 255) / 256, dh, B_);
          hipLaunchKernelGGL(gather_nk_kernel, g, dim3(256), 0, stream, fagg, dh, 0, nidx[i], fn, dh, 0, N); }
        // att2 fc, split input [fn | fxyz2(ovl)]
        conv(E.a2fc, nullptr, nullptr, d, d, fn, dh, 0, dh, ovl, dh, 0,
             nullptr, 0, 0, attb, d, 0, NK, 0);
        { dim3 g((N + 255) / 256, d, B_);
          hipLaunchKernelGGL(softmax_agg_kernel, g, dim3(256), 0, stream, attb, fn, dh, ovl, aggb, d, N); }
        // att2 mlp: d -> d
        conv(E.a2m.w, E.a2m.g, E.a2m.bta, d, d, aggb, d, 0, d,
             nullptr, 0, 0, nullptr, 0, 0, f2b, d, 0, N, 1);
        // shortcut: din -> 2d (no act)
        conv(E.sc.w, E.sc.g, E.sc.bta, 2 * d, din, fin, din, 0, din,
             nullptr, 0, 0, nullptr, 0, 0, scb, 2 * d, 0, N, 0);
        // mlp2: d -> 2d, + shortcut residual, leaky
        conv(E.m2.w, E.m2.g, E.m2.bta, 2 * d, d, f2b, d, 0, d,
             nullptr, 0, 0, scb, 2 * d, 0, feb, 2 * d, 0, N, 1);
        // random_sample (max over gathered neighbours)
        { dim3 g((MS[i] + 255) / 256, 2 * d, B_);
          hipLaunchKernelGGL(maxpool_kernel, g, dim3(256), 0, stream, feb, sidx[i], skipP[i], 2 * d, N, MS[i]); }
    }

    // ----------------- decoder -----------------
    size_t so = sbase;
    auto sal = [&](size_t nf) { float* p = ws + so; so += nf; return p; };
    float* da  = sal((size_t)B_ * 64 * 65536);
    float* db  = sal((size_t)B_ * 64 * 65536);
    float* dfi = sal((size_t)4194304);           // max B*Cprev*Nj over decoder

    // dec0: 1024 -> 1024 on 128 pts
    conv(dec0W.w, dec0W.g, dec0W.bta, 1024, 1024, skipP[4], 1024, 0, 1024,
         nullptr, 0, 0, nullptr, 0, 0, da, 1024, 0, MS[4], 1);

    float* fcur = da; float* fnx = db;
    int Cprev = 1024, prevN = MS[4];
    static const int DCO[5] = {512, 256, 128, 32, 32};
    for (int j = 0; j < 5; ++j) {
        const int Nj = NS[4 - j];
        { dim3 g((Nj + 255) / 256, Cprev, B_);
          hipLaunchKernelGGL(interp_kernel, g, dim3(256), 0, stream, fcur, iidx[4 - j], dfi, Cprev, prevN, Nj); }
        const float* skip = (j < 4) ? skipP[3 - j] : skipFE0;
        const int Cskip   = (j < 4) ? 2 * OC[3 - j] : 32;
        conv(decW[j].w, decW[j].g, decW[j].bta, DCO[j], Cskip + Cprev,
             skip, Cskip, 0, Cskip, dfi, Cprev, 0,
             nullptr, 0, 0, fnx, DCO[j], 0, Nj, 1);
        float* tmp = fcur; fcur = fnx; fnx = tmp;
        Cprev = DCO[j]; prevN = Nj;
    }

    // fc1: 32 -> 64, fc2: 64 -> 32, fc3: 32 -> 13 (+bias, no BN/act)
    conv(fc1W.w, fc1W.g, fc1W.bta, 64, 32, fcur, 32, 0, 32,
         nullptr, 0, 0, nullptr, 0, 0, fnx, 64, 0, NS[0], 1);
    conv(fc2W.w, fc2W.g, fc2W.bta, 32, 64, fnx, 64, 0, 64,
         nullptr, 0, 0, nullptr, 0, 0, fcur, 32, 0, NS[0], 1);
    conv(fc3w, nullptr, fc3b, 13, 32, fcur, 32, 0, 32,
         nullptr, 0, 0, nullptr, 0, 0, (float*)d_out, 13, 0, NS[0], 0);
}